// GINEVirtualNodeClassifier_3934190044111
// MI455X (gfx1250) — compile-verified
//
#include <hip/hip_runtime.h>
#include <hip/hip_bf16.h>

typedef __attribute__((ext_vector_type(2))) float v2f;
typedef __attribute__((ext_vector_type(8))) float v8f;

#define NND 50000
#define NE  800000
#define HID 128
#define NG  128

// ---------------------------------------------------------------------------
// Generic fp32 WMMA GEMM: C[M,128] = op(A[M,K] @ B[K,128] + bias)
// flags: bit0 = relu, bit1 = accumulate into C, bit2 = emit column sum/sumsq
// Block = 256 threads (8 waves). Each block owns 16 rows; wave w owns cols
// [16w,16w+16). Uses V_WMMA_F32_16X16X4_F32; A strip staged in padded LDS.
// ---------------------------------------------------------------------------
__global__ __launch_bounds__(256) void gemm_wmma_kernel(
    const float* __restrict__ A, const float* __restrict__ B,
    const float* __restrict__ bias, float* __restrict__ C,
    int K, int flags, float* __restrict__ colsum, float* __restrict__ colsq)
{
    __shared__ float As[16 * 132];          // K<=128, row pad +4 (bank spread)
    const int KP  = K + 4;
    const int tid = threadIdx.x;
    const int m0  = blockIdx.x * 16;

    for (int i = tid; i < 16 * K; i += 256) {
        int r = i / K, c = i - r * K;
        As[r * KP + c] = A[(size_t)(m0 + r) * K + c];
    }
    __syncthreads();

    const int wave  = tid >> 5;
    const int lane  = tid & 31;
    const int lhalf = lane >> 4;            // 0: K pair {0,1}, 1: {2,3}
    const int lm    = lane & 15;
    const int n     = wave * 16 + lm;       // output column

    v8f acc = {};
    const float* As_row = &As[lm * KP];
    for (int k0 = 0; k0 < K; k0 += 4) {
        int ka = k0 + 2 * lhalf;
        v2f a, b;
        a.x = As_row[ka];
        a.y = As_row[ka + 1];
        b.x = B[(size_t)ka * HID + n];
        b.y = B[(size_t)(ka + 1) * HID + n];
        acc = __builtin_amdgcn_wmma_f32_16x16x4_f32(
            false, a, false, b, (short)0, acc, false, false);
    }

    const float bv = bias[n];
    float csum = 0.f, csq = 0.f;
    const int mrow = m0 + 8 * lhalf;        // C/D layout: vgpr v -> row v + 8*lhalf
#pragma unroll
    for (int v = 0; v < 8; ++v) {
        float val = acc[v] + bv;
        if (flags & 1) val = fmaxf(val, 0.f);
        float* cp = C + (size_t)(mrow + v) * HID + n;
        if (flags & 2) val += *cp;
        *cp = val;
        csum += val;
        csq  += val * val;
    }
    if (flags & 4) {
        csum += __shfl_xor(csum, 16);       // combine row-halves (same column)
        csq  += __shfl_xor(csq, 16);
        if (lhalf == 0) {
            unsafeAtomicAdd(&colsum[n], csum);
            unsafeAtomicAdd(&colsq[n],  csq);
        }
    }
}

// h[n] += vn[batch[n]]  (in place), and z[n] = h[n]  (aggregation base)
__global__ __launch_bounds__(256) void addvn_kernel(
    float* __restrict__ h, const float* __restrict__ vn,
    const int* __restrict__ batch, float* __restrict__ z, int N)
{
    int t = blockIdx.x * blockDim.x + threadIdx.x;
    if (t >= N * 32) return;
    int node = t >> 5;
    int c    = (t & 31) * 4;
    int g    = batch[node];
    float4 hv = *(const float4*)&h[(size_t)node * HID + c];
    float4 vv = *(const float4*)&vn[(size_t)g * HID + c];
    hv.x += vv.x; hv.y += vv.y; hv.z += vv.z; hv.w += vv.w;
    *(float4*)&h[(size_t)node * HID + c] = hv;
    *(float4*)&z[(size_t)node * HID + c] = hv;
}

// per edge: m = relu(h[src] + edge_attr@W_e + b_e);  z[dst] += m (fp32 atomics)
__global__ __launch_bounds__(256) void msg_kernel(
    const float* __restrict__ h, const float* __restrict__ edge_attr,
    const int* __restrict__ src, const int* __restrict__ dst,
    const float* __restrict__ W_e, const float* __restrict__ b_e,
    float* __restrict__ z, int E)
{
    int t = blockIdx.x * blockDim.x + threadIdx.x;
    int eid = t >> 5;
    if (eid >= E) return;
    int c  = (t & 31) * 4;
    float a0 = edge_attr[(size_t)eid * 2 + 0];
    float a1 = edge_attr[(size_t)eid * 2 + 1];
    int s = src[eid], d = dst[eid];
    float4 hv = *(const float4*)&h[(size_t)s * HID + c];
#pragma unroll
    for (int j = 0; j < 4; ++j) {
        float e  = a0 * W_e[c + j] + a1 * W_e[HID + c + j] + b_e[c + j];
        float hj = (&hv.x)[j];
        float m  = fmaxf(hj + e, 0.f);
        unsafeAtomicAdd(&z[(size_t)d * HID + c + j], m);
    }
}

// BatchNorm (batch stats) + relu + optional global_add_pool into vn_up
__global__ __launch_bounds__(256) void bn_kernel(
    const float* __restrict__ z, float* __restrict__ h,
    const float* __restrict__ csum, const float* __restrict__ csq,
    const float* __restrict__ gamma, const float* __restrict__ beta,
    const int* __restrict__ batch, float* __restrict__ vn_up,
    int do_pool, int N)
{
    int idx = blockIdx.x * blockDim.x + threadIdx.x;
    if (idx >= N * HID) return;
    int node = idx >> 7, c = idx & 127;
    float invN = 1.0f / (float)N;
    float mu   = csum[c] * invN;
    float var  = csq[c] * invN - mu * mu;
    float val  = (z[idx] - mu) * rsqrtf(var + 1e-5f) * gamma[c] + beta[c];
    val = fmaxf(val, 0.f);
    h[idx] = val;
    if (do_pool)
        unsafeAtomicAdd(&vn_up[(size_t)batch[node] * HID + c], val);
}

__global__ __launch_bounds__(256) void pool_kernel(
    const float* __restrict__ h, const int* __restrict__ batch,
    float* __restrict__ gsum, float* __restrict__ cnt, int N)
{
    int idx = blockIdx.x * blockDim.x + threadIdx.x;
    if (idx >= N * HID) return;
    int node = idx >> 7, c = idx & 127;
    int g = batch[node];
    unsafeAtomicAdd(&gsum[(size_t)g * HID + c], h[idx]);
    if (c == 0) unsafeAtomicAdd(&cnt[g], 1.0f);
}

__global__ __launch_bounds__(256) void gdiv_kernel(
    const float* __restrict__ gsum, const float* __restrict__ cnt,
    float* __restrict__ g)
{
    int idx = blockIdx.x * blockDim.x + threadIdx.x;
    if (idx >= NG * HID) return;
    int r = idx >> 7;
    g[idx] = gsum[idx] / fmaxf(cnt[r], 1.0f);
}

__global__ __launch_bounds__(128) void logits_kernel(
    const float* __restrict__ t, const float* __restrict__ W2,
    const float* __restrict__ b2, float* __restrict__ out)
{
    __shared__ float red[128];
    int g = blockIdx.x, c = threadIdx.x;
    red[c] = t[(size_t)g * HID + c] * W2[c];
    __syncthreads();
    for (int s = 64; s > 0; s >>= 1) {
        if (c < s) red[c] += red[c + s];
        __syncthreads();
    }
    if (c == 0) out[g] = red[0] + b2[0];
}

__global__ void zero_kernel(float* __restrict__ p, int n)
{
    int i = blockIdx.x * blockDim.x + threadIdx.x;
    if (i < n) p[i] = 0.f;
}

extern "C" void kernel_launch(void* const* d_in, const int* in_sizes, int n_in,
                              void* d_out, int out_size, void* d_ws, size_t ws_size,
                              hipStream_t stream)
{
    (void)in_sizes; (void)n_in; (void)out_size; (void)ws_size;
    const float* x         = (const float*)d_in[0];
    const float* edge_attr = (const float*)d_in[1];
    const int*   ei        = (const int*)  d_in[2];
    const int*   batch     = (const int*)  d_in[3];
    const float* W_in      = (const float*)d_in[4];
    const float* b_in      = (const float*)d_in[5];
    const float* W_e       = (const float*)d_in[6];
    const float* b_e       = (const float*)d_in[7];
    const float* conv_W1   = (const float*)d_in[8];
    const float* conv_b1   = (const float*)d_in[9];
    const float* conv_W2   = (const float*)d_in[10];
    const float* conv_b2   = (const float*)d_in[11];
    const float* bn_g      = (const float*)d_in[12];
    const float* bn_b      = (const float*)d_in[13];
    const float* vn_W1     = (const float*)d_in[14];
    const float* vn_b1     = (const float*)d_in[15];
    const float* vn_W2     = (const float*)d_in[16];
    const float* vn_b2     = (const float*)d_in[17];
    const float* cls_W1    = (const float*)d_in[18];
    const float* cls_b1    = (const float*)d_in[19];
    const float* cls_W2    = (const float*)d_in[20];
    const float* cls_b2    = (const float*)d_in[21];

    char* ws = (char*)d_ws;
    size_t off = 0;
    auto alloc = [&](size_t bytes) -> float* {
        float* p = (float*)(ws + off);
        off += (bytes + 255) & ~(size_t)255;
        return p;
    };
    const int N = NND, E = NE, H = HID, G = NG;
    float* hbuf   = alloc((size_t)N * H * 4);
    float* zbuf   = alloc((size_t)N * H * 4);
    float* tbuf   = alloc((size_t)N * H * 4);
    float* vn     = alloc((size_t)G * H * 4);
    float* vn_up  = alloc((size_t)G * H * 4);
    float* tmp128 = alloc((size_t)G * H * 4);
    float* gsum   = alloc((size_t)G * H * 4);
    float* gbuf   = alloc((size_t)G * H * 4);
    float* csum   = alloc(H * 4);   // csum,csq adjacent (512B each, contiguous)
    float* csq    = alloc(H * 4);
    float* cnt    = alloc(G * 4);

    dim3 blk(256);
    auto grid1 = [](int n) { return dim3((unsigned)((n + 255) / 256)); };

    // init: vn = 0 ; h = x @ W_in + b_in
    zero_kernel<<<grid1(G * H), blk, 0, stream>>>(vn, G * H);
    gemm_wmma_kernel<<<dim3(N / 16), blk, 0, stream>>>(
        x, W_in, b_in, hbuf, 64, 0, nullptr, nullptr);

    for (int i = 0; i < 4; ++i) {
        // h += vn[batch] ; z = h
        addvn_kernel<<<grid1(N * 32), blk, 0, stream>>>(hbuf, vn, batch, zbuf, N);
        // z[dst] += relu(h[src] + edge_attr@W_e + b_e)
        msg_kernel<<<grid1(E * 32), blk, 0, stream>>>(
            hbuf, edge_attr, ei, ei + E, W_e, b_e, zbuf, E);
        // t = relu(z @ W1 + b1)
        gemm_wmma_kernel<<<dim3(N / 16), blk, 0, stream>>>(
            zbuf, conv_W1 + (size_t)i * H * H, conv_b1 + i * H, tbuf, H, 1,
            nullptr, nullptr);
        // z = t @ W2 + b2, with fused column sum/sumsq for BN
        zero_kernel<<<dim3(1), blk, 0, stream>>>(csum, 2 * H);
        gemm_wmma_kernel<<<dim3(N / 16), blk, 0, stream>>>(
            tbuf, conv_W2 + (size_t)i * H * H, conv_b2 + i * H, zbuf, H, 4,
            csum, csq);
        if (i < 3)
            zero_kernel<<<grid1(G * H), blk, 0, stream>>>(vn_up, G * H);
        // h = relu(BN(z)); pool into vn_up when needed
        bn_kernel<<<grid1(N * H), blk, 0, stream>>>(
            zbuf, hbuf, csum, csq, bn_g + i * H, bn_b + i * H, batch, vn_up,
            (i < 3) ? 1 : 0, N);
        if (i < 3) {
            // vn += relu(vn_up @ vn_W1 + b1) @ vn_W2 + b2
            gemm_wmma_kernel<<<dim3(G / 16), blk, 0, stream>>>(
                vn_up, vn_W1, vn_b1, tmp128, H, 1, nullptr, nullptr);
            gemm_wmma_kernel<<<dim3(G / 16), blk, 0, stream>>>(
                tmp128, vn_W2, vn_b2, vn, H, 2, nullptr, nullptr);
        }
    }

    // global mean pool + classifier
    zero_kernel<<<grid1(G * H), blk, 0, stream>>>(gsum, G * H);
    zero_kernel<<<dim3(1), dim3(128), 0, stream>>>(cnt, G);
    pool_kernel<<<grid1(N * H), blk, 0, stream>>>(hbuf, batch, gsum, cnt, N);
    gdiv_kernel<<<grid1(G * H), blk, 0, stream>>>(gsum, cnt, gbuf);
    gemm_wmma_kernel<<<dim3(G / 16), blk, 0, stream>>>(
        gbuf, cls_W1, cls_b1, tmp128, H, 1, nullptr, nullptr);
    logits_kernel<<<dim3(G), dim3(128), 0, stream>>>(
        tmp128, cls_W2, cls_b2, (float*)d_out);
}